// XLMRForTokenClassification_46205258170741
// MI455X (gfx1250) — compile-verified
//
#include <hip/hip_runtime.h>

// ---------------------------------------------------------------------------
// XLM-R token-classification loss, fused for MI455X (gfx1250, wave32).
// Memory-bound: 256 MB of activations -> ~11us floor at 23.3 TB/s.
// Head GEMM via V_WMMA_F32_16X16X4_F32 (16 tokens x 16 classes per wave tile).
// Inner loop is pure load+WMMA: invalid (past-n_valid) rows are fixed up in the
// epilogue (their true logits are exactly `bias`), not masked per-load.
// ---------------------------------------------------------------------------

typedef float v2f __attribute__((ext_vector_type(2)));
typedef float v8f __attribute__((ext_vector_type(8)));

#define BDIM 128
#define SDIM 512
#define HDIM 1024
#define LDIM 9
#define WROWS 10       // 9 real rows + 1 zero row for lanes n >= 9
#define WSTRIDE 1028   // padded row stride (n*1028 % 64 distinct for n<16) -> conflict-free

// --------------------------------------------------------------------------
// Kernel 1: per-batch stable compaction map.
// srcOf[b*S + j] = original position of j-th valid token, or -1 if j >= n_valid.
// --------------------------------------------------------------------------
__global__ __launch_bounds__(SDIM) void k_compact(const int* __restrict__ bpe,
                                                  int* __restrict__ srcOf) {
    __shared__ int sc[SDIM];
    const int b = blockIdx.x;
    const int t = threadIdx.x;
    const int flag = (bpe[b * SDIM + t] == 1) ? 1 : 0;
    sc[t] = flag;
    __syncthreads();
    // Hillis-Steele inclusive scan over 512 entries.
    for (int off = 1; off < SDIM; off <<= 1) {
        int v = (t >= off) ? sc[t - off] : 0;
        __syncthreads();
        sc[t] += v;
        __syncthreads();
    }
    const int total = sc[SDIM - 1];
    const int excl  = sc[t] - flag;
    if (flag)       srcOf[b * SDIM + excl] = t;   // dest slots [0, total)
    if (t >= total) srcOf[b * SDIM + t]    = -1;  // dest slots [total, S) -> zero rows
}

// --------------------------------------------------------------------------
// Kernel 2: fused head GEMM (WMMA f32 16x16x4) + log-softmax + masked NLL.
// 512 blocks x 256 threads; each wave32 owns one 16-token tile.
// --------------------------------------------------------------------------
__global__ __launch_bounds__(256) void k_main(const float* __restrict__ X,
                                              const int*   __restrict__ labels,
                                              const int*   __restrict__ lmask,
                                              const float* __restrict__ W,
                                              const float* __restrict__ bias,
                                              const int*   __restrict__ srcOf,
                                              float*       __restrict__ partials) {
    __shared__ float ldsW[WROWS * WSTRIDE];  // 41.1 KB: W rows 0..8, row 9 = zeros
    __shared__ float biasS[16];
    __shared__ float tile[8][16][16];        // per-wave logits tile
    __shared__ float redS[256];
    __shared__ float redC[256];

    const int tid = threadIdx.x;

    // Stage W into LDS (coalesced); row 9 is a zero row for padded N lanes.
    for (int i = tid; i < WROWS * HDIM; i += 256) {
        const int r = i >> 10;          // / HDIM
        const int c = i & (HDIM - 1);   // % HDIM
        ldsW[r * WSTRIDE + c] = (r < LDIM) ? W[i] : 0.0f;
    }
    if (tid < 16) biasS[tid] = (tid < LDIM) ? bias[tid] : 0.0f;
    __syncthreads();

    const int wave   = tid >> 5;
    const int lane   = tid & 31;
    const int tileId = blockIdx.x * 8 + wave;      // 0..4095
    const int bidx   = tileId >> 5;                // 32 tiles per batch row
    const int jt     = (tileId & 31) << 4;         // first dest position of tile

    // A-matrix (16x4 f32) lane mapping: lanes 0-15 -> M, lane-half selects K pair.
    const int   m      = lane & 15;
    const int   khalf  = (lane >> 4) << 1;         // 0 or 2
    const int   src    = srcOf[bidx * SDIM + jt + m];
    // Invalid rows load safely from row 0; their accumulator rows are garbage
    // and get replaced by `bias` in the epilogue (0-row x W + b == b).
    const float* rowp  = X + ((size_t)bidx * SDIM + (src >= 0 ? src : 0)) * HDIM + khalf;

    // B-matrix (4x16 f32) lane mapping mirrors A; lanes n>=9 read the zero row.
    const int   n     = lane & 15;
    const int   nrow  = (n < LDIM) ? n : LDIM;     // broadcast zero row, no conflicts
    const float* wrow = &ldsW[nrow * WSTRIDE + khalf];

    v8f acc0 = {}, acc1 = {}, acc2 = {}, acc3 = {};

    // Software-pipelined, fully branchless K loop (prefetch next 16-wide chunk).
    v2f a0 = *(const v2f*)(rowp +  0);
    v2f a1 = *(const v2f*)(rowp +  4);
    v2f a2 = *(const v2f*)(rowp +  8);
    v2f a3 = *(const v2f*)(rowp + 12);
    v2f b0 = *(const v2f*)(wrow +  0);
    v2f b1 = *(const v2f*)(wrow +  4);
    v2f b2 = *(const v2f*)(wrow +  8);
    v2f b3 = *(const v2f*)(wrow + 12);

    for (int k0 = 0; k0 < HDIM; k0 += 16) {
        const int kn = (k0 + 16) & (HDIM - 1);     // wraps to 0 on last iteration
        v2f na0 = *(const v2f*)(rowp + kn +  0);
        v2f na1 = *(const v2f*)(rowp + kn +  4);
        v2f na2 = *(const v2f*)(rowp + kn +  8);
        v2f na3 = *(const v2f*)(rowp + kn + 12);
        v2f nb0 = *(const v2f*)(wrow + kn +  0);
        v2f nb1 = *(const v2f*)(wrow + kn +  4);
        v2f nb2 = *(const v2f*)(wrow + kn +  8);
        v2f nb3 = *(const v2f*)(wrow + kn + 12);

        acc0 = __builtin_amdgcn_wmma_f32_16x16x4_f32(false, a0, false, b0, (short)0, acc0, false, false);
        acc1 = __builtin_amdgcn_wmma_f32_16x16x4_f32(false, a1, false, b1, (short)0, acc1, false, false);
        acc2 = __builtin_amdgcn_wmma_f32_16x16x4_f32(false, a2, false, b2, (short)0, acc2, false, false);
        acc3 = __builtin_amdgcn_wmma_f32_16x16x4_f32(false, a3, false, b3, (short)0, acc3, false, false);

        a0 = na0; a1 = na1; a2 = na2; a3 = na3;
        b0 = nb0; b1 = nb1; b2 = nb2; b3 = nb3;
    }
    v8f acc = (acc0 + acc1) + (acc2 + acc3);

    // C/D layout: lane holds column n; VGPR i -> row (lane<16 ? i : 8+i).
    const float bn    = (n < LDIM) ? biasS[n] : 0.0f;
    const int   mbase = (lane >> 4) << 3;
#pragma unroll
    for (int i = 0; i < 8; ++i)
        tile[wave][mbase + i][n] = acc[i] + bn;
    __syncthreads();

    // Lanes 0-15: log-softmax over 9 classes + masked NLL for row m == lane.
    // For lanes 0-15, `src` above was loaded for exactly this row.
    float myNll = 0.0f, myCnt = 0.0f;
    if (lane < 16) {
        const bool ok = (src >= 0);
        const int j = jt + lane;
        float x[LDIM];
#pragma unroll
        for (int l = 0; l < LDIM; ++l)
            x[l] = ok ? tile[wave][lane][l] : biasS[l];
        float mx = x[0];
#pragma unroll
        for (int l = 1; l < LDIM; ++l) mx = fmaxf(mx, x[l]);
        float s = 0.0f;
#pragma unroll
        for (int l = 0; l < LDIM; ++l) s += __expf(x[l] - mx);
        const float lse = mx + __logf(s);
        const int lab = labels[bidx * SDIM + j];
        const int msk = lmask[bidx * SDIM + j];
        if (msk == 1 && lab != 0) {
            myNll = lse - x[lab];
            myCnt = 1.0f;
        }
    }
    redS[tid] = myNll;
    redC[tid] = myCnt;
    __syncthreads();
    for (int off = 128; off > 0; off >>= 1) {
        if (tid < off) {
            redS[tid] += redS[tid + off];
            redC[tid] += redC[tid + off];
        }
        __syncthreads();
    }
    if (tid == 0) {
        partials[blockIdx.x * 2 + 0] = redS[0];
        partials[blockIdx.x * 2 + 1] = redC[0];
    }
}

// --------------------------------------------------------------------------
// Kernel 3: deterministic final reduction over 512 block partials.
// --------------------------------------------------------------------------
__global__ void k_final(const float* __restrict__ partials, float* __restrict__ out) {
    if (threadIdx.x == 0 && blockIdx.x == 0) {
        float s = 0.0f, c = 0.0f;
        for (int i = 0; i < 512; ++i) {
            s += partials[2 * i + 0];
            c += partials[2 * i + 1];
        }
        out[0] = s / fmaxf(c, 1.0f);
    }
}

// --------------------------------------------------------------------------
extern "C" void kernel_launch(void* const* d_in, const int* in_sizes, int n_in,
                              void* d_out, int out_size, void* d_ws, size_t ws_size,
                              hipStream_t stream) {
    const float* X      = (const float*)d_in[0];  // [B,S,H] f32
    const int*   labels = (const int*)  d_in[1];  // [B,S]
    const int*   lmask  = (const int*)  d_in[2];  // [B,S]
    const int*   bpe    = (const int*)  d_in[3];  // [B,S]
    const float* W      = (const float*)d_in[4];  // [L,H]
    const float* bias   = (const float*)d_in[5];  // [L]
    float*       out    = (float*)d_out;

    int*   srcOf    = (int*)d_ws;                                       // B*S ints (256 KB)
    float* partials = (float*)((char*)d_ws + (size_t)BDIM * SDIM * 4);  // 512*2 floats

    k_compact<<<BDIM, SDIM, 0, stream>>>(bpe, srcOf);
    k_main<<<512, 256, 0, stream>>>(X, labels, lmask, W, bias, srcOf, partials);
    k_final<<<1, 64, 0, stream>>>(partials, out);
}